// BEV_corr_65274912965337
// MI455X (gfx1250) — compile-verified
//
#include <hip/hip_runtime.h>
#include <stdint.h>

// Problem constants (from reference)
#define BB 16
#define CC 256
#define HH 64
#define WW 256
#define SS 64
#define NPIX (SS * SS)        // 4096 BEV pixels
#define SLICE (HH * WW)       // 16384 f32 elements = 64 KB per (b,c) channel slice
#define MASK_OFF ((size_t)BB * CC * NPIX)

typedef __attribute__((ext_vector_type(4))) unsigned int v4u_t;
typedef __attribute__((ext_vector_type(8))) int v8i_t;
typedef __attribute__((ext_vector_type(4))) int v4i_t;

#if __has_builtin(__builtin_amdgcn_tensor_load_to_lds)
#define HAVE_TDM 1
#endif

__device__ __forceinline__ float pymodf(float a, float m) {
  // Python-style modulo: result in [0, m)
  float r = fmodf(a, m);
  return (r < 0.0f) ? r + m : r;
}

// ---------------------------------------------------------------------------
// Kernel 1: per-pixel sampling coordinates, bilinear weights, packed corner
// indices (u16 pairs) and the mask output. Done once per (b, pixel) instead of
// once per (b, c, pixel) — 256x less transcendental work.
// ---------------------------------------------------------------------------
__global__ void bev_coords_kernel(const float* __restrict__ rot,
                                  const float* __restrict__ shu,
                                  const float* __restrict__ shv,
                                  const float* __restrict__ mpp,
                                  float4* __restrict__ wgt,
                                  uint2* __restrict__ idx,
                                  float* __restrict__ mask_out) {
  const float TWO_PI = 6.2831853071795864769f;
  const float PI = 3.14159265358979323846f;

  int t = blockIdx.x * blockDim.x + threadIdx.x;  // 0 .. B*NPIX-1
  int b = t >> 12;                                // NPIX == 4096
  int p = t & (NPIX - 1);
  int i = p >> 6;                                 // row in S x S grid
  int j = p & (SS - 1);                           // col

  float cv = 31.5f + shv[b];                      // S/2 - 0.5 + shift_v
  float cu = 31.5f + shu[b];
  float dy = (float)i - cv;
  float dx = (float)j - cu;

  float radius = sqrtf(dy * dy + dx * dx);
  float th = atan2f(dy, dx);                      // [-pi, pi]
  th = pymodf(th, TWO_PI);
  th = pymodf(th - 0.5f * PI, TWO_PI);            // (-pi/2 + theta%2pi) % 2pi
  th = pymodf(th + rot[b] * TWO_PI, TWO_PI);      // rot * 360/180*pi = rot*2pi
  float ix = th * ((float)WW / TWO_PI);           // -> x in [0, W)

  float iy = atan2f(radius * (mpp[b] * 8.0f), -2.0f) * ((float)HH / PI);

  // Bilinear prep (matches reference: clamp corners first, weights from
  // clamped corners, zero-mask outside image)
  float x0 = floorf(ix), y0 = floorf(iy);
  float x1 = x0 + 1.0f, y1 = y0 + 1.0f;
  float m = (ix >= 0.0f && ix <= (float)(WW - 1) &&
             iy >= 0.0f && iy <= (float)(HH - 1)) ? 1.0f : 0.0f;

  float cx0 = fminf(fmaxf(x0, 0.0f), (float)(WW - 1));
  float cx1 = fminf(fmaxf(x1, 0.0f), (float)(WW - 1));
  float cy0 = fminf(fmaxf(y0, 0.0f), (float)(HH - 1));
  float cy1 = fminf(fmaxf(y1, 0.0f), (float)(HH - 1));

  float wnw = (cx1 - ix) * (cy1 - iy) * m;
  float wne = (ix - cx0) * (cy1 - iy) * m;
  float wsw = (cx1 - ix) * (iy - cy0) * m;
  float wse = (ix - cx0) * (iy - cy0) * m;

  int inw = (int)cy0 * WW + (int)cx0;
  int ine = (int)cy0 * WW + (int)cx1;
  int isw = (int)cy1 * WW + (int)cx0;
  int ise = (int)cy1 * WW + (int)cx1;

  wgt[t] = make_float4(wnw, wne, wsw, wse);
  idx[t] = make_uint2((unsigned)inw | ((unsigned)ine << 16),
                      (unsigned)isw | ((unsigned)ise << 16));
  mask_out[t] = m;
}

// ---------------------------------------------------------------------------
// TDM issue: DMA one 64KB channel slice (contiguous, 16384 x f32) from global
// to LDS. D# per CDNA5 ISA ch.8: group0 = {count=1, lds_addr, global_addr,
// type=2}; group1 = {data_size=4B, tensor_dim0=16384, tensor_dim1=1,
// tile_dim0=16384, tile_dim1/2 unused}; groups 2/3 zero (<=2D tensor).
// No wait here — caller pipelines with s_wait_tensorcnt.
// ---------------------------------------------------------------------------
#ifdef HAVE_TDM
__device__ __forceinline__ void tdm_issue_slice(const float* gsrc, unsigned lds_off) {
  uint64_t ga = (uint64_t)(uintptr_t)gsrc;
  v4u_t g0;
  g0.x = 1u;                                            // count=1, user mode
  g0.y = lds_off;                                       // lds_addr (bytes)
  g0.z = (unsigned)(ga & 0xffffffffu);                  // global_addr[31:0]
  g0.w = (unsigned)((ga >> 32) & 0x01ffffffu) | (2u << 30);  // addr[56:32], type=2
  v8i_t g1;
  g1[0] = 0x00020000;                                   // data_size=2 -> 4 bytes
  g1[1] = (int)(16384u << 16);                          // tensor_dim0[15:0] @ [31:16]
  g1[2] = (int)(1u << 16);                              // tensor_dim0 hi=0; tensor_dim1=1
  g1[3] = (int)(16384u << 16);                          // tensor_dim1 hi=0; tile_dim0=16384
  g1[4] = 0;                                            // tile_dim1=0, tile_dim2=0 (unused)
  g1[5] = 16384;                                        // tensor_dim0_stride lo
  g1[6] = 0;
  g1[7] = 0;
  v4i_t g2 = (v4i_t)0;
  v4i_t g3 = (v4i_t)0;
#if defined(__clang_major__) && (__clang_major__ >= 23)
  v8i_t g4 = (v8i_t)0;
  __builtin_amdgcn_tensor_load_to_lds(g0, g1, g2, g3, g4, 0);
#else
  __builtin_amdgcn_tensor_load_to_lds(g0, g1, g2, g3, 0);
#endif
}
#endif

// ---------------------------------------------------------------------------
// Kernel 2: one workgroup per (b, channel-pair). Double-buffered TDM staging:
// wave 0 issues both 64KB slice DMAs back-to-back; block computes channel c0
// from buffer 0 while the TDM engine streams channel c1 into buffer 1
// (same-wave tensor ops complete in order, so tensorcnt<=1 means buf0 ready,
// tensorcnt<=0 means buf1 ready). 256 threads (8 wave32s), 128KB LDS
// (2 blocks/WGP of the 320KB pool).
// ---------------------------------------------------------------------------
__global__ void bev_sample_kernel(const float* __restrict__ grd,
                                  const float4* __restrict__ wgt,
                                  const uint2* __restrict__ idx,
                                  float* __restrict__ out) {
  extern __shared__ float tile[];                  // 2 * SLICE floats = 128 KB
  float* buf0 = tile;
  float* buf1 = tile + SLICE;

  int b = blockIdx.x >> 7;                         // CC/2 == 128 pairs per batch
  const float* src0 = grd + (size_t)(blockIdx.x * 2) * SLICE;
  const float* src1 = src0 + SLICE;

#ifdef HAVE_TDM
  if (threadIdx.x < 32) {                          // wave 0 issues both DMAs
    unsigned off0 = (unsigned)(uintptr_t)(void*)buf0;
    unsigned off1 = (unsigned)(uintptr_t)(void*)buf1;
    tdm_issue_slice(src0, off0);
    tdm_issue_slice(src1, off1);
    __builtin_amdgcn_s_wait_tensorcnt(1);          // buf0 complete (in-order)
  }
#else
  // Fallback: cooperative coalesced fill (also taken by the host pass)
  for (int e = threadIdx.x; e < (2 * SLICE) / 4; e += blockDim.x) {
    ((float4*)tile)[e] = ((const float4*)src0)[e];
  }
#endif
  __syncthreads();

  const float4* wb = wgt + b * NPIX;
  const uint2* ib = idx + b * NPIX;
  float* ob0 = out + (size_t)(blockIdx.x * 2) * NPIX;
  float* ob1 = ob0 + NPIX;

  // Channel c0 from buf0 — overlaps with the in-flight DMA of c1 into buf1.
  for (int p = threadIdx.x; p < NPIX; p += blockDim.x) {
    float4 w = wb[p];
    uint2 id = ib[p];
    float v = w.x * buf0[id.x & 0xffffu]
            + w.y * buf0[id.x >> 16]
            + w.z * buf0[id.y & 0xffffu]
            + w.w * buf0[id.y >> 16];
    __builtin_nontemporal_store(v, &ob0[p]);       // streamed, never re-read
  }

#ifdef HAVE_TDM
  if (threadIdx.x < 32) {
    __builtin_amdgcn_s_wait_tensorcnt(0);          // buf1 complete
  }
#endif
  __syncthreads();

  // Channel c1 from buf1.
  for (int p = threadIdx.x; p < NPIX; p += blockDim.x) {
    float4 w = wb[p];
    uint2 id = ib[p];
    float v = w.x * buf1[id.x & 0xffffu]
            + w.y * buf1[id.x >> 16]
            + w.z * buf1[id.y & 0xffffu]
            + w.w * buf1[id.y >> 16];
    __builtin_nontemporal_store(v, &ob1[p]);
  }
}

extern "C" void kernel_launch(void* const* d_in, const int* in_sizes, int n_in,
                              void* d_out, int out_size, void* d_ws, size_t ws_size,
                              hipStream_t stream) {
  const float* grd = (const float*)d_in[0];   // [16,256,64,256]
  const float* rot = (const float*)d_in[1];   // [16]
  const float* shu = (const float*)d_in[2];   // [16]
  const float* shv = (const float*)d_in[3];   // [16]
  const float* mpp = (const float*)d_in[4];   // [16]

  float* out = (float*)d_out;                                   // [16,256,64,64]
  float* mask_out = out + MASK_OFF;                             // [16,1,64,64]

  float4* wgt = (float4*)d_ws;                                  // 65536 * 16B = 1 MB
  uint2* idx = (uint2*)((char*)d_ws + (size_t)BB * NPIX * sizeof(float4));  // +512 KB

  bev_coords_kernel<<<(BB * NPIX) / 256, 256, 0, stream>>>(
      rot, shu, shv, mpp, wgt, idx, mask_out);

  bev_sample_kernel<<<(BB * CC) / 2, 256, 2 * SLICE * sizeof(float), stream>>>(
      grd, wgt, idx, out);
}